// VectorQuantizer_8598524526680
// MI455X (gfx1250) — compile-verified
//
#include <hip/hip_runtime.h>

typedef float v2f __attribute__((ext_vector_type(2)));
typedef float v8f __attribute__((ext_vector_type(8)));

#define HEADS   4
#define DIM     32
#define KCODES  64
#define HD      128      // HEADS*DIM
#define WAVES   4
#define TPB     128      // WAVES * 32
#define XP_STRIDE 132    // 128 + 4 pad -> conflict-free column reads

__device__ __forceinline__ v8f wmma_f32x4(v2f a, v2f b, v8f c) {
    // D = A(16x4 f32) * B(4x16 f32) + C(16x16 f32)
    return __builtin_amdgcn_wmma_f32_16x16x4_f32(
        false, a, false, b, (short)0, c, false, false);
}

__global__ void __launch_bounds__(TPB) vq_main_kernel(
    const float* __restrict__ x,      // [N,32]
    const float* __restrict__ w_in,   // [32,128]
    const float* __restrict__ b_in,   // [128]
    const float* __restrict__ embed,  // [4,64,32]
    const float* __restrict__ w_out,  // [128,32]
    const float* __restrict__ b_out,  // [32]
    float* __restrict__ out,          // [N,32]
    float* __restrict__ partial,      // [gridDim.x] loss partials
    int nTiles)
{
    __shared__ float xpSh[WAVES][16 * XP_STRIDE]; // per-wave xp tile [16][128]
    __shared__ float e2Sh[HEADS * KCODES];        // codebook squared norms
    __shared__ int   idxSh[WAVES][HEADS][16];     // per-row argmin indices
    __shared__ float lossSh[TPB];

    const int tid  = threadIdx.x;
    const int wave = tid >> 5;
    const int lane = tid & 31;
    const int n    = lane & 15;       // row-in-tile for A / column for B,C
    const int half = lane >> 4;
    const int koff = half * 2;        // K-pair selected by lane half

    // ---- precompute e2[h][k] = sum_d embed[h][k][d]^2 ----
    for (int c = tid; c < HEADS * KCODES; c += TPB) {
        const float* e = embed + (size_t)c * DIM;
        float s = 0.f;
        #pragma unroll
        for (int d = 0; d < DIM; ++d) s += e[d] * e[d];
        e2Sh[c] = s;
    }
    __syncthreads();

    long tile = (long)blockIdx.x * WAVES + wave;
    const bool active = (tile < nTiles);
    if (!active) tile = 0;            // clamp so idle waves stay barrier-uniform
    const long base = tile * 16;

    float* xpW = xpSh[wave];

    // =============== GEMM1: xp = x @ w_in + b_in  (16x128) ===============
    {
        const float* xrow = x + (base + n) * DIM;
        v2f a[8];
        #pragma unroll
        for (int k = 0; k < 8; ++k)
            a[k] = *(const v2f*)(xrow + 4 * k + koff);

        #pragma unroll
        for (int j = 0; j < 8; ++j) {
            v8f acc = {0.f, 0.f, 0.f, 0.f, 0.f, 0.f, 0.f, 0.f};
            #pragma unroll
            for (int k = 0; k < 8; ++k) {
                v2f b;
                b.x = w_in[(4 * k + koff)     * HD + 16 * j + n];
                b.y = w_in[(4 * k + koff + 1) * HD + 16 * j + n];
                acc = wmma_f32x4(a[k], b, acc);
            }
            const float bi = b_in[16 * j + n];
            #pragma unroll
            for (int r = 0; r < 8; ++r) {
                const float v = acc[r] + bi;
                xpW[(r + 8 * half) * XP_STRIDE + 16 * j + n] = v;
            }
        }
    }
    __syncthreads();

    // =============== distances + argmin per head =========================
    #pragma unroll
    for (int h = 0; h < HEADS; ++h) {
        // B fragments: B[kk][n] = xp[n][32h + 4k + kk]  (hoisted over code tiles)
        v2f bb[8];
        #pragma unroll
        for (int k = 0; k < 8; ++k)
            bb[k] = *(const v2f*)(xpW + n * XP_STRIDE + h * DIM + 4 * k + koff);

        float bs = 3.0e38f;
        int   bi = 0;
        #pragma unroll
        for (int t = 0; t < 4; ++t) {          // 16-code tiles
            v8f cc = {0.f, 0.f, 0.f, 0.f, 0.f, 0.f, 0.f, 0.f};
            const float* erow = embed + ((size_t)(h * KCODES + 16 * t + n)) * DIM;
            #pragma unroll
            for (int k = 0; k < 8; ++k) {
                v2f ea = *(const v2f*)(erow + 4 * k + koff);
                cc = wmma_f32x4(ea, bb[k], cc);  // xe^T: M=code, N=row
            }
            #pragma unroll
            for (int r = 0; r < 8; ++r) {
                const int code = 16 * t + r + 8 * half;
                const float sc = e2Sh[h * KCODES + code] - 2.0f * cc[r];
                if (sc < bs) { bs = sc; bi = code; }   // ascending scan -> first-min
            }
        }
        // combine the two lane-halves holding the same row n
        const float os = __shfl_xor(bs, 16, 32);
        const int   oi = __shfl_xor(bi, 16, 32);
        if (os < bs || (os == bs && oi < bi)) { bs = os; bi = oi; }
        if (half == 0) idxSh[wave][h][n] = bi;
    }
    __syncthreads();

    // =============== GEMM2: out = quant @ w_out + b_out, + commit loss ===
    int myIdx[HEADS];
    #pragma unroll
    for (int h = 0; h < HEADS; ++h) myIdx[h] = idxSh[wave][h][n];

    v8f o0 = {0.f, 0.f, 0.f, 0.f, 0.f, 0.f, 0.f, 0.f};
    v8f o1 = {0.f, 0.f, 0.f, 0.f, 0.f, 0.f, 0.f, 0.f};
    float lacc = 0.f;

    #pragma unroll
    for (int k = 0; k < 32; ++k) {             // K over 128
        const int h  = k >> 3;
        const int kk = (k & 7) * 4 + koff;     // within-head d offset
        const float* qrow = embed + ((size_t)(h * KCODES + myIdx[h])) * DIM;
        const v2f qa = *(const v2f*)(qrow + kk);
        const v2f xv = *(const v2f*)(xpW + n * XP_STRIDE + h * DIM + kk);
        const float dx = qa.x - xv.x, dy = qa.y - xv.y;
        lacc += dx * dx + dy * dy;

        const int kr = 4 * k + koff;           // row in w_out
        v2f b0, b1;
        b0.x = w_out[kr * DIM + n];        b0.y = w_out[(kr + 1) * DIM + n];
        b1.x = w_out[kr * DIM + 16 + n];   b1.y = w_out[(kr + 1) * DIM + 16 + n];
        o0 = wmma_f32x4(qa, b0, o0);
        o1 = wmma_f32x4(qa, b1, o1);
    }

    if (active) {
        const float bo0 = b_out[n], bo1 = b_out[16 + n];
        #pragma unroll
        for (int r = 0; r < 8; ++r) {
            const long m = base + r + 8 * half;
            out[m * DIM + n]      = o0[r] + bo0;
            out[m * DIM + 16 + n] = o1[r] + bo1;
        }
    }

    // ---- deterministic block loss reduction ----
    lossSh[tid] = active ? lacc : 0.f;
    __syncthreads();
    #pragma unroll
    for (int s = TPB / 2; s > 0; s >>= 1) {
        if (tid < s) lossSh[tid] += lossSh[tid + s];
        __syncthreads();
    }
    if (tid == 0) partial[blockIdx.x] = lossSh[0];
}

__global__ void __launch_bounds__(256) vq_loss_reduce_kernel(
    const float* __restrict__ partial, int nb, float* __restrict__ lossOut,
    float scale)
{
    __shared__ float sh[256];
    const int tid = threadIdx.x;
    float s = 0.f;
    for (int i = tid; i < nb; i += 256) s += partial[i];  // fixed order
    sh[tid] = s;
    __syncthreads();
    #pragma unroll
    for (int k = 128; k > 0; k >>= 1) {
        if (tid < k) sh[tid] += sh[tid + k];
        __syncthreads();
    }
    if (tid == 0) *lossOut = sh[0] * scale;
}

extern "C" void kernel_launch(void* const* d_in, const int* in_sizes, int n_in,
                              void* d_out, int out_size, void* d_ws, size_t ws_size,
                              hipStream_t stream) {
    const float* x     = (const float*)d_in[0];
    const float* w_in  = (const float*)d_in[1];
    const float* b_in  = (const float*)d_in[2];
    const float* embed = (const float*)d_in[3];
    const float* w_out = (const float*)d_in[4];
    const float* b_out = (const float*)d_in[5];

    const int N = in_sizes[0] / DIM;          // 1,000,000
    const int nTiles = N / 16;                // N is a multiple of 16 here
    const int blocks = (nTiles + WAVES - 1) / WAVES;

    float* out     = (float*)d_out;           // [N,32] then 1 loss scalar
    float* lossOut = out + (size_t)N * DIM;
    float* partial = (float*)d_ws;            // blocks floats

    vq_main_kernel<<<blocks, TPB, 0, stream>>>(
        x, w_in, b_in, embed, w_out, b_out, out, partial, nTiles);

    const float scale = 1.0f / ((float)((size_t)N * HEADS * DIM)); // COMMIT_W=1
    vq_loss_reduce_kernel<<<1, 256, 0, stream>>>(partial, blocks, lossOut, scale);
}